// Detection_2920577761704
// MI455X (gfx1250) — compile-verified
//
#include <hip/hip_runtime.h>
#include <stdint.h>

#define NBOX   2048
#define NT     256
#define NCLS   5
#define NKEEP  4                 // foreground classes (softmax cols 1..4)
#define WORDS  (NBOX / 32)       // 64 bitmask words per row
#define CHUNK  64                // sorted rows per mask chunk (16 KB LDS)
#define NCHUNK (NBOX / CHUNK)
#define OVERLAP_C 0.5f
#define THRESH_C  0.3f

typedef __attribute__((ext_vector_type(2))) float v2f;
typedef __attribute__((ext_vector_type(8))) float v8f;
typedef __attribute__((address_space(3))) void lvoid;

#if __has_builtin(__builtin_amdgcn_wmma_f32_16x16x4_f32)
#define HAVE_WMMA4 1
#else
#define HAVE_WMMA4 0
#endif

__global__ __launch_bounds__(NT) void detect_nms_kernel(
    const float* __restrict__ loc,   // (8,2048,2)
    const float* __restrict__ cls,   // (8,2048,5)
    const float* __restrict__ defs,  // (2048,2)
    float* __restrict__ out)         // (8,4,2048,3)
{
  __shared__ float          sk[NBOX];      // sorted scores (keys)
  __shared__ unsigned short si[NBOX];      // sorted original indices
  __shared__ float          dstart[NBOX];  // decoded, original order
  __shared__ float          dend[NBOX];
  __shared__ float          ss[NBOX];      // decoded, sorted order
  __shared__ float          se[NBOX];
  __shared__ __attribute__((aligned(16))) unsigned smask[CHUNK * WORDS]; // 16 KB
  __shared__ unsigned       kw[WORDS];     // final removed-bitmask

  const int tid  = threadIdx.x;
  const int lane = tid & 31;
  const int wid  = tid >> 5;
  const int bb   = blockIdx.x >> 2;  // batch
  const int cc   = blockIdx.x & 3;   // foreground class index (0..3 -> col cc+1)

  // ---- stage localizations_default into LDS (aliased onto smask region) ----
  // Async DMA (gfx1250): global_load_async_to_lds_b128, tracked by ASYNCcnt.
  float* sDef = (float*)smask; // 4096 floats == 16 KB, exactly smask's size
  {
    const unsigned ldsbase = (unsigned)(uintptr_t)(lvoid*)sDef;
#pragma unroll
    for (int k = 0; k < (NBOX * 2) / (NT * 4); ++k) { // 4 iters, 16B per lane
      unsigned long long ga =
          (unsigned long long)(uintptr_t)(defs + 4 * (tid + k * NT));
      unsigned la = ldsbase + 16u * (unsigned)(tid + k * NT);
      asm volatile("global_load_async_to_lds_b128 %0, %1, off"
                   :: "v"(la), "v"(ga) : "memory");
    }
    asm volatile("s_wait_asynccnt 0" ::: "memory");
  }
  __syncthreads();

  // ---- softmax (class cc+1) + box decode ----
  __builtin_prefetch(cls + ((size_t)bb * NBOX + tid) * NCLS, 0, 1); // global_prefetch_b8
  for (int n = tid; n < NBOX; n += NT) {
    const float* cp = cls + ((size_t)bb * NBOX + n) * NCLS;
    float l0 = cp[0], l1 = cp[1], l2 = cp[2], l3 = cp[3], l4 = cp[4];
    float m  = fmaxf(fmaxf(fmaxf(l0, l1), fmaxf(l2, l3)), l4);
    float e0 = __expf(l0 - m), e1 = __expf(l1 - m), e2 = __expf(l2 - m);
    float e3 = __expf(l3 - m), e4 = __expf(l4 - m);
    float denom = e0 + e1 + e2 + e3 + e4;
    float num = (cc == 0) ? e1 : (cc == 1) ? e2 : (cc == 2) ? e3 : e4;
    sk[n] = num / denom;
    si[n] = (unsigned short)n;

    const float* lp2 = loc + ((size_t)bb * NBOX + n) * 2;
    float c0 = sDef[2 * n], w0 = sDef[2 * n + 1];
    float cen = c0 + lp2[0] * w0;
    float wdt = w0 * __expf(lp2[1]);
    dstart[n] = cen - 0.5f * wdt;
    dend[n]   = cen + 0.5f * wdt;
  }
  __syncthreads();

  // ---- bitonic sort by score, descending (payload = original index) ----
  for (int k = 2; k <= NBOX; k <<= 1) {
    for (int j = k >> 1; j > 0; j >>= 1) {
      __syncthreads();
      for (int p = tid; p < NBOX; p += NT) {
        int q = p ^ j;
        if (q > p) {
          bool descBlk = ((p & k) == 0);
          float a = sk[p], b = sk[q];
          bool sw = descBlk ? (a < b) : (a > b);
          if (sw) {
            sk[p] = b; sk[q] = a;
            unsigned short t = si[p]; si[p] = si[q]; si[q] = t;
          }
        }
      }
    }
  }
  __syncthreads();

  // ---- gather boxes into sorted order ----
  for (int p = tid; p < NBOX; p += NT) {
    int n = si[p];
    ss[p] = dstart[n];
    se[p] = dend[n];
  }
  __syncthreads();

  // ---- chunked mask build (WMMA tiles) + sequential greedy scan ----
  unsigned rem_lo = 0u, rem_hi = 0u; // wave0: lane owns removed words [lane] and [lane+32]
  for (int chunk = 0; chunk < NCHUNK; ++chunk) {
    const int base = chunk * CHUNK;
    for (int i = tid; i < CHUNK * WORDS; i += NT) smask[i] = 0u;
    __syncthreads();

    int cnt = 0;
    for (int tr = 0; tr < CHUNK / 16; ++tr) {
      const int tig = (base >> 4) + tr;
      const int i0  = tig << 4;
      for (int tj = tig; tj < NBOX / 16; ++tj, ++cnt) {
        if ((cnt & 7) != wid) continue; // wave-uniform tile distribution
        const int j0  = tj << 4;
        const bool lo = lane < 16;
        const int l15 = lane & 15;
        const float siv = ss[i0 + l15];
        const float eiv = se[i0 + l15];
        const float sjv = ss[j0 + l15];
        const float ejv = se[j0 + l15];
#if HAVE_WMMA4
        // A row i = [s_i, e_i, 1, 0]; lanes<16 hold K=0,1, lanes>=16 hold K=2,3
        v2f A;   A.x   = lo ? siv  : 1.0f; A.y   = lo ? eiv  : 0.0f;
        v2f Bsp; Bsp.x = lo ? 1.0f : 0.0f; Bsp.y = lo ? sjv  : 0.0f; // -> s_i + s_j
        v2f Bsm; Bsm.x = Bsp.x;            Bsm.y = lo ? -sjv : 0.0f; // -> s_i - s_j
        v2f Bep; Bep.x = lo ? 0.0f : 1.0f; Bep.y = lo ? ejv  : 0.0f; // -> e_i + e_j
        v2f Bem; Bem.x = Bep.x;            Bem.y = lo ? -ejv : 0.0f; // -> e_i - e_j
        v8f z = {};
        v8f Ssum = __builtin_amdgcn_wmma_f32_16x16x4_f32(false, A, false, Bsp, (short)0, z, false, false);
        v8f Sdif = __builtin_amdgcn_wmma_f32_16x16x4_f32(false, A, false, Bsm, (short)0, z, false, false);
        v8f Esum = __builtin_amdgcn_wmma_f32_16x16x4_f32(false, A, false, Bep, (short)0, z, false, false);
        v8f Edif = __builtin_amdgcn_wmma_f32_16x16x4_f32(false, A, false, Bem, (short)0, z, false, false);
#endif
#pragma unroll
        for (int r = 0; r < 8; ++r) {
          const int ig = i0 + r + (lo ? 0 : 8); // C/D layout: VGPR r -> M=r (lo), M=r+8 (hi)
          const int jg = j0 + l15;
#if HAVE_WMMA4
          float ssum = Ssum[r], sdif = Sdif[r], esum = Esum[r], edif = Edif[r];
#else
          float si2 = ss[ig], ei2 = se[ig];
          float ssum = si2 + sjv, sdif = si2 - sjv;
          float esum = ei2 + ejv, edif = ei2 - ejv;
#endif
          // inter = min(e_i,e_j) - max(s_i,s_j), clipped at 0
          float inter = fmaxf(0.5f * ((esum - fabsf(edif)) - (ssum + fabsf(sdif))), 0.0f);
          float uni   = fmaxf((esum - ssum) - inter, 1e-12f);
          bool pred = (inter > OVERLAP_C * uni) && (jg > ig);
          unsigned m = (unsigned)__ballot(pred);
          const unsigned lo16 = m & 0xffffu; // rows M=r    , cols j0..j0+15
          const unsigned hi16 = m >> 16;     // rows M=r+8  , cols j0..j0+15
          const int wsel = j0 >> 5;
          const int sh   = j0 & 16;
          if (lane == 0  && lo16) atomicOr(&smask[(i0 - base + r    ) * WORDS + wsel], lo16 << sh);
          if (lane == 16 && hi16) atomicOr(&smask[(i0 - base + r + 8) * WORDS + wsel], hi16 << sh);
        }
      }
    }
    __syncthreads();

    // greedy scan over this chunk's sorted rows, wave 0 only
    if (tid < 32) {
      for (int ir = 0; ir < CHUNK; ++ir) {
        const int ig = base + ir;
        const int w  = ig >> 5;                       // word index 0..63 (uniform)
        const unsigned cur = (w < 32) ? rem_lo : rem_hi;
        const unsigned wv  = __shfl(cur, w & 31, 32);
        const bool removed = (wv >> (ig & 31)) & 1u;
        const bool valid   = sk[ig] > THRESH_C;
        if (valid && !removed) {
          rem_lo |= smask[ir * WORDS + lane];
          rem_hi |= smask[ir * WORDS + 32 + lane];
        }
      }
    }
    __syncthreads();
  }
  if (tid < 32) { kw[tid] = rem_lo; kw[tid + 32] = rem_hi; }
  __syncthreads();

  // ---- write output: (b, cc, n, {start,end,score}) zeroed when not kept ----
  for (int p = tid; p < NBOX; p += NT) {
    bool removed = (kw[p >> 5] >> (p & 31)) & 1u;
    float s0 = ss[p], e0 = se[p], sc = sk[p];
    bool keep = (sc > THRESH_C) && !removed && (s0 > -10.0f) && (e0 < 10.0f);
    int n = si[p];
    size_t o = ((size_t)(bb * NKEEP + cc) * NBOX + n) * 3;
    out[o + 0] = keep ? s0 : 0.0f;
    out[o + 1] = keep ? e0 : 0.0f;
    out[o + 2] = keep ? sc : 0.0f;
  }
}

extern "C" void kernel_launch(void* const* d_in, const int* in_sizes, int n_in,
                              void* d_out, int out_size, void* d_ws, size_t ws_size,
                              hipStream_t stream) {
  (void)in_sizes; (void)n_in; (void)d_ws; (void)ws_size; (void)out_size;
  const float* loc  = (const float*)d_in[0];
  const float* cls  = (const float*)d_in[1];
  const float* defs = (const float*)d_in[2];
  float* out = (float*)d_out;
  dim3 grid(8 * NKEEP), block(NT);
  detect_nms_kernel<<<grid, block, 0, stream>>>(loc, cls, defs, out);
}